// OLMoESparseMoE_74036646248878
// MI455X (gfx1250) — compile-verified
//
#include <hip/hip_runtime.h>
#include <math.h>

// ---------------- problem constants ----------------
#define D_    1024
#define DFF_  4096
#define E_    8
#define NTOK  8192      // B*S = 4*2048

typedef float v2f __attribute__((ext_vector_type(2)));
typedef float v8f __attribute__((ext_vector_type(8)));

// fp32 WMMA: D(16x16 f32) = A(16x4 f32) * B(4x16 f32) + C
__device__ __forceinline__ v8f wmma_f32_k4(v2f a, v2f b, v8f c) {
    return __builtin_amdgcn_wmma_f32_16x16x4_f32(
        /*neg_a=*/false, a, /*neg_b=*/false, b,
        /*c_mod=*/(short)0, c, /*reuse_a=*/false, /*reuse_b=*/false);
}

__device__ __forceinline__ float gelu_exact(float v) {
    return 0.5f * v * (1.0f + erff(v * 0.70710678118654752440f));
}

// ---------------- kernel 0: zero out + counters ----------------
__global__ void moe_zero_kernel(float* __restrict__ out, long long n,
                                int* __restrict__ counts) {
    long long i = (long long)blockIdx.x * blockDim.x + threadIdx.x;
    const long long stride = (long long)gridDim.x * blockDim.x;
    for (; i < n; i += stride) out[i] = 0.0f;
    if (blockIdx.x == 0 && threadIdx.x < E_) counts[threadIdx.x] = 0;
}

// ---------------- kernel 1: gating (one wave32 per token) ----------------
__global__ __launch_bounds__(256) void moe_gate_kernel(
    const float* __restrict__ x, const float* __restrict__ Wg,
    float* __restrict__ gates, int* __restrict__ counts,
    int* __restrict__ lists) {
    const int lane = threadIdx.x & 31;
    const int wave = threadIdx.x >> 5;
    const int t = blockIdx.x * 8 + wave;
    if (t >= NTOK) return;

    float s[E_];
#pragma unroll
    for (int e = 0; e < E_; ++e) s[e] = 0.0f;
    const float* xr = x + (size_t)t * D_;
    for (int d = lane; d < D_; d += 32) {
        const float xv = xr[d];
#pragma unroll
        for (int e = 0; e < E_; ++e) s[e] += xv * Wg[e * D_ + d];
    }
#pragma unroll
    for (int e = 0; e < E_; ++e) {
#pragma unroll
        for (int off = 16; off > 0; off >>= 1)
            s[e] += __shfl_xor(s[e], off, 32);
    }
    if (lane == 0) {
        // top-2 (ties -> lowest index first, matching lax.top_k)
        int i0 = 0; float v0 = s[0];
#pragma unroll
        for (int e = 1; e < E_; ++e) if (s[e] > v0) { v0 = s[e]; i0 = e; }
        int i1 = -1; float v1 = -3.4e38f;
#pragma unroll
        for (int e = 0; e < E_; ++e)
            if (e != i0 && s[e] > v1) { v1 = s[e]; i1 = e; }
        // softmax over the two selected logits (v0 >= v1)
        const float p1 = expf(v1 - v0);
        const float inv = 1.0f / (1.0f + p1);
        const float g0 = inv, g1 = p1 * inv;
#pragma unroll
        for (int e = 0; e < E_; ++e) gates[t * E_ + e] = 0.0f;
        gates[t * E_ + i0] = g0;
        gates[t * E_ + i1] = g1;
        const int p = atomicAdd(&counts[i0], 1); lists[i0 * NTOK + p] = t;
        const int q = atomicAdd(&counts[i1], 1); lists[i1 * NTOK + q] = t;
    }
}

// ---------------- kernel 2: load-balancing loss (deterministic) ----------------
__global__ __launch_bounds__(256) void moe_loss_kernel(
    const float* __restrict__ gates, float* __restrict__ loss_out) {
    __shared__ float part[256];
    const int e = threadIdx.x & 7;
    const int slot = threadIdx.x >> 3;   // 0..31
    float s = 0.0f;
    for (int t = slot; t < NTOK; t += 32) s += gates[t * E_ + e];
    part[threadIdx.x] = s;               // tid == slot*8 + e
    __syncthreads();
    if (threadIdx.x == 0) {
        float u[E_];
        float mu = 0.0f;
        for (int ee = 0; ee < E_; ++ee) {
            float acc = 0.0f;
            for (int sl = 0; sl < 32; ++sl) acc += part[sl * 8 + ee];
            u[ee] = acc / (float)NTOK;
            mu += u[ee];
        }
        mu *= (1.0f / (float)E_);
        float var = 0.0f;
        for (int ee = 0; ee < E_; ++ee) {
            const float d = u[ee] - mu;
            var += d * d;
        }
        var *= (1.0f / (float)(E_ - 1));   // ddof=1
        loss_out[0] = var * 0.01f;
    }
}

// =====================================================================
// GEMM tiling shared by fc1/fc2:
//   block tile 64(M) x 128(N), Kc=32, 256 threads = 8 waves (2 M x 4 N),
//   each wave computes 32x32 via 4 accumulators; A and B^T both stored
//   in LDS with even stride 34 so every fragment is one ds_load_b64.
// =====================================================================
#define KC_   32
#define LDSA_ 34
#define LDSB_ 34

// ---------------- kernel 3: fc1  H = gelu(X_gathered @ W1e + b1e) ----------------
__global__ __launch_bounds__(256) void moe_fc1_kernel(
    const float* __restrict__ x, const float* __restrict__ W1e,
    const float* __restrict__ b1e, const int* __restrict__ lists,
    const int* __restrict__ counts, float* __restrict__ H,
    int expert, int chunkBase, int Tc) {
    const int cnt = counts[expert];
    int rowsInChunk = cnt - chunkBase;
    if (rowsInChunk > Tc) rowsInChunk = Tc;
    const int mBase = blockIdx.x * 64;
    if (rowsInChunk <= mBase) return;     // uniform early-exit
    const int nBase = blockIdx.y * 128;

    __shared__ float As [64][LDSA_];      // [m][k]
    __shared__ float Bst[128][LDSB_];     // [n][k]  (transposed)

    const int tid  = threadIdx.x;
    const int lane = tid & 31, wave = tid >> 5;
    const int half = lane >> 4, lm = lane & 15;
    const int Moff = 32 * (wave & 1);     // {0,32}
    const int Noff = 32 * (wave >> 1);    // {0,32,64,96}

    // A staging: 64 rows x 32 k = 2 float4 per thread
    const int rA0 = tid >> 3, rA1 = rA0 + 32;
    const int cA  = (tid & 7) << 2;
    const int* lst = lists + expert * NTOK + chunkBase;
    const bool va0 = (mBase + rA0) < rowsInChunk;
    const bool va1 = (mBase + rA1) < rowsInChunk;
    const float* ap0 = va0 ? (x + (size_t)lst[mBase + rA0] * D_) : x;
    const float* ap1 = va1 ? (x + (size_t)lst[mBase + rA1] * D_) : x;

    // B staging: 32 k-rows x 128 n = 4 float4 per thread (write transposed)
    const int rB0 = tid >> 4, rB1 = rB0 + 16;   // k rows
    const int cB0 = (tid & 15) << 2, cB1 = cB0 + 64;

    v8f acc00 = {}; v8f acc01 = {}; v8f acc10 = {}; v8f acc11 = {};

    for (int k0 = 0; k0 < D_; k0 += KC_) {
        if (va0) {
            const float4 q = *(const float4*)(ap0 + k0 + cA);
            As[rA0][cA] = q.x; As[rA0][cA+1] = q.y; As[rA0][cA+2] = q.z; As[rA0][cA+3] = q.w;
        } else {
            As[rA0][cA] = 0.f; As[rA0][cA+1] = 0.f; As[rA0][cA+2] = 0.f; As[rA0][cA+3] = 0.f;
        }
        if (va1) {
            const float4 q = *(const float4*)(ap1 + k0 + cA);
            As[rA1][cA] = q.x; As[rA1][cA+1] = q.y; As[rA1][cA+2] = q.z; As[rA1][cA+3] = q.w;
        } else {
            As[rA1][cA] = 0.f; As[rA1][cA+1] = 0.f; As[rA1][cA+2] = 0.f; As[rA1][cA+3] = 0.f;
        }
        {
            const float4 q = *(const float4*)(W1e + (size_t)(k0 + rB0) * DFF_ + nBase + cB0);
            Bst[cB0+0][rB0] = q.x; Bst[cB0+1][rB0] = q.y; Bst[cB0+2][rB0] = q.z; Bst[cB0+3][rB0] = q.w;
        }
        {
            const float4 q = *(const float4*)(W1e + (size_t)(k0 + rB0) * DFF_ + nBase + cB1);
            Bst[cB1+0][rB0] = q.x; Bst[cB1+1][rB0] = q.y; Bst[cB1+2][rB0] = q.z; Bst[cB1+3][rB0] = q.w;
        }
        {
            const float4 q = *(const float4*)(W1e + (size_t)(k0 + rB1) * DFF_ + nBase + cB0);
            Bst[cB0+0][rB1] = q.x; Bst[cB0+1][rB1] = q.y; Bst[cB0+2][rB1] = q.z; Bst[cB0+3][rB1] = q.w;
        }
        {
            const float4 q = *(const float4*)(W1e + (size_t)(k0 + rB1) * DFF_ + nBase + cB1);
            Bst[cB0+64][rB1] = q.x; Bst[cB1+1][rB1] = q.y; Bst[cB1+2][rB1] = q.z; Bst[cB1+3][rB1] = q.w;
        }
        __syncthreads();
#pragma unroll
        for (int kk = 0; kk < KC_; kk += 4) {
            const int ka = kk + 2 * half;
            v2f a0; a0.x = As[Moff + lm][ka];       a0.y = As[Moff + lm][ka + 1];
            v2f a1; a1.x = As[Moff + 16 + lm][ka];  a1.y = As[Moff + 16 + lm][ka + 1];
            v2f b0; b0.x = Bst[Noff + lm][ka];      b0.y = Bst[Noff + lm][ka + 1];
            v2f b1; b1.x = Bst[Noff + 16 + lm][ka]; b1.y = Bst[Noff + 16 + lm][ka + 1];
            acc00 = wmma_f32_k4(a0, b0, acc00);
            acc01 = wmma_f32_k4(a0, b1, acc01);
            acc10 = wmma_f32_k4(a1, b0, acc10);
            acc11 = wmma_f32_k4(a1, b1, acc11);
        }
        __syncthreads();
    }
#pragma unroll
    for (int r = 0; r < 8; ++r) {
        const int m0 = Moff + r + 8 * half;
        const int lrow0 = mBase + m0, lrow1 = lrow0 + 16;
        const int n0 = nBase + Noff + lm, n1 = n0 + 16;
        if (lrow0 < rowsInChunk) {
            H[(size_t)lrow0 * DFF_ + n0] = gelu_exact(acc00[r] + b1e[n0]);
            H[(size_t)lrow0 * DFF_ + n1] = gelu_exact(acc01[r] + b1e[n1]);
        }
        if (lrow1 < rowsInChunk) {
            H[(size_t)lrow1 * DFF_ + n0] = gelu_exact(acc10[r] + b1e[n0]);
            H[(size_t)lrow1 * DFF_ + n1] = gelu_exact(acc11[r] + b1e[n1]);
        }
    }
}

// ---------------- kernel 4: fc2  out[tok] += gate * (H @ W2e + b2e) ----------------
__global__ __launch_bounds__(256) void moe_fc2_kernel(
    const float* __restrict__ H, const float* __restrict__ W2e,
    const float* __restrict__ b2e, const int* __restrict__ lists,
    const int* __restrict__ counts, const float* __restrict__ gates,
    float* __restrict__ out, int expert, int chunkBase, int Tc) {
    const int cnt = counts[expert];
    int rowsInChunk = cnt - chunkBase;
    if (rowsInChunk > Tc) rowsInChunk = Tc;
    const int mBase = blockIdx.x * 64;
    if (rowsInChunk <= mBase) return;
    const int nBase = blockIdx.y * 128;

    __shared__ float As [64][LDSA_];
    __shared__ float Bst[128][LDSB_];

    const int tid  = threadIdx.x;
    const int lane = tid & 31, wave = tid >> 5;
    const int half = lane >> 4, lm = lane & 15;
    const int Moff = 32 * (wave & 1);
    const int Noff = 32 * (wave >> 1);

    const int rA0 = tid >> 3, rA1 = rA0 + 32;
    const int cA  = (tid & 7) << 2;
    const bool va0 = (mBase + rA0) < rowsInChunk;
    const bool va1 = (mBase + rA1) < rowsInChunk;
    const float* ap0 = H + (size_t)(mBase + rA0) * DFF_;
    const float* ap1 = H + (size_t)(mBase + rA1) * DFF_;

    const int rB0 = tid >> 4, rB1 = rB0 + 16;
    const int cB0 = (tid & 15) << 2, cB1 = cB0 + 64;

    v8f acc00 = {}; v8f acc01 = {}; v8f acc10 = {}; v8f acc11 = {};

    for (int k0 = 0; k0 < DFF_; k0 += KC_) {
        if (va0) {
            const float4 q = *(const float4*)(ap0 + k0 + cA);
            As[rA0][cA] = q.x; As[rA0][cA+1] = q.y; As[rA0][cA+2] = q.z; As[rA0][cA+3] = q.w;
        } else {
            As[rA0][cA] = 0.f; As[rA0][cA+1] = 0.f; As[rA0][cA+2] = 0.f; As[rA0][cA+3] = 0.f;
        }
        if (va1) {
            const float4 q = *(const float4*)(ap1 + k0 + cA);
            As[rA1][cA] = q.x; As[rA1][cA+1] = q.y; As[rA1][cA+2] = q.z; As[rA1][cA+3] = q.w;
        } else {
            As[rA1][cA] = 0.f; As[rA1][cA+1] = 0.f; As[rA1][cA+2] = 0.f; As[rA1][cA+3] = 0.f;
        }
        {
            const float4 q = *(const float4*)(W2e + (size_t)(k0 + rB0) * D_ + nBase + cB0);
            Bst[cB0+0][rB0] = q.x; Bst[cB0+1][rB0] = q.y; Bst[cB0+2][rB0] = q.z; Bst[cB0+3][rB0] = q.w;
        }
        {
            const float4 q = *(const float4*)(W2e + (size_t)(k0 + rB0) * D_ + nBase + cB1);
            Bst[cB1+0][rB0] = q.x; Bst[cB1+1][rB0] = q.y; Bst[cB1+2][rB0] = q.z; Bst[cB1+3][rB0] = q.w;
        }
        {
            const float4 q = *(const float4*)(W2e + (size_t)(k0 + rB1) * D_ + nBase + cB0);
            Bst[cB0+0][rB1] = q.x; Bst[cB0+1][rB1] = q.y; Bst[cB0+2][rB1] = q.z; Bst[cB0+3][rB1] = q.w;
        }
        {
            const float4 q = *(const float4*)(W2e + (size_t)(k0 + rB1) * D_ + nBase + cB1);
            Bst[cB1+0][rB1] = q.x; Bst[cB1+1][rB1] = q.y; Bst[cB1+2][rB1] = q.z; Bst[cB1+3][rB1] = q.w;
        }
        __syncthreads();
#pragma unroll
        for (int kk = 0; kk < KC_; kk += 4) {
            const int ka = kk + 2 * half;
            v2f a0; a0.x = As[Moff + lm][ka];       a0.y = As[Moff + lm][ka + 1];
            v2f a1; a1.x = As[Moff + 16 + lm][ka];  a1.y = As[Moff + 16 + lm][ka + 1];
            v2f b0; b0.x = Bst[Noff + lm][ka];      b0.y = Bst[Noff + lm][ka + 1];
            v2f b1; b1.x = Bst[Noff + 16 + lm][ka]; b1.y = Bst[Noff + 16 + lm][ka + 1];
            acc00 = wmma_f32_k4(a0, b0, acc00);
            acc01 = wmma_f32_k4(a0, b1, acc01);
            acc10 = wmma_f32_k4(a1, b0, acc10);
            acc11 = wmma_f32_k4(a1, b1, acc11);
        }
        __syncthreads();
    }
    const int* lst = lists + expert * NTOK + chunkBase;
#pragma unroll
    for (int r = 0; r < 8; ++r) {
        const int m0 = Moff + r + 8 * half;
        const int lrow0 = mBase + m0, lrow1 = lrow0 + 16;
        const int n0 = nBase + Noff + lm, n1 = n0 + 16;
        if (lrow0 < rowsInChunk) {
            const int tok = lst[lrow0];
            const float g = gates[tok * E_ + expert];
            out[(size_t)tok * D_ + n0] += g * (acc00[r] + b2e[n0]);
            out[(size_t)tok * D_ + n1] += g * (acc01[r] + b2e[n1]);
        }
        if (lrow1 < rowsInChunk) {
            const int tok = lst[lrow1];
            const float g = gates[tok * E_ + expert];
            out[(size_t)tok * D_ + n0] += g * (acc10[r] + b2e[n0]);
            out[(size_t)tok * D_ + n1] += g * (acc11[r] + b2e[n1]);
        }
    }
}

// ---------------- host launcher ----------------
extern "C" void kernel_launch(void* const* d_in, const int* in_sizes, int n_in,
                              void* d_out, int out_size, void* d_ws, size_t ws_size,
                              hipStream_t stream) {
    const float* x  = (const float*)d_in[0];
    const float* Wg = (const float*)d_in[1];
    const float* W1 = (const float*)d_in[2];
    const float* b1 = (const float*)d_in[3];
    const float* W2 = (const float*)d_in[4];
    const float* b2 = (const float*)d_in[5];
    float* out = (float*)d_out;

    char* ws = (char*)d_ws;
    int*   counts = (int*)ws;                                  // 8 ints
    int*   lists  = (int*)(ws + 256);                          // E*NTOK ints
    float* gates  = (float*)(ws + 256 + 4ull * E_ * NTOK);     // NTOK*E floats
    size_t OFS_H  = 256 + 8ull * E_ * NTOK;
    OFS_H = (OFS_H + 255) & ~(size_t)255;
    float* H = (float*)(ws + OFS_H);

    // token chunk so H (Tc x DFF fp32) fits in workspace
    size_t avail = (ws_size > OFS_H) ? (ws_size - OFS_H) : 0;
    long long tcll = (long long)(avail / ((size_t)DFF_ * 4));
    int Tc = (int)(tcll > NTOK ? NTOK : tcll);
    Tc &= ~63;
    if (Tc <= 0) Tc = 64;
    const int chunks = (NTOK + Tc - 1) / Tc;

    const long long outN = (long long)NTOK * D_ + 1;
    moe_zero_kernel<<<2048, 256, 0, stream>>>(out, outN, counts);
    moe_gate_kernel<<<NTOK / 8, 256, 0, stream>>>(x, Wg, gates, counts, lists);
    moe_loss_kernel<<<1, 256, 0, stream>>>(gates, out + (outN - 1));

    for (int e = 0; e < E_; ++e) {
        const float* W1e = W1 + (size_t)e * D_ * DFF_;
        const float* b1e = b1 + (size_t)e * DFF_;
        const float* W2e = W2 + (size_t)e * DFF_ * D_;
        const float* b2e = b2 + (size_t)e * D_;
        for (int c = 0; c < chunks; ++c) {
            dim3 g1(Tc / 64, DFF_ / 128);
            moe_fc1_kernel<<<g1, 256, 0, stream>>>(x, W1e, b1e, lists, counts,
                                                   H, e, c * Tc, Tc);
            dim3 g2(Tc / 64, D_ / 128);
            moe_fc2_kernel<<<g2, 256, 0, stream>>>(H, W2e, b2e, lists, counts,
                                                   gates, out, e, c * Tc, Tc);
        }
    }
}